// GAT_55551107007265
// MI455X (gfx1250) — compile-verified
//
#include <hip/hip_runtime.h>

typedef __attribute__((ext_vector_type(2))) float v2f;
typedef __attribute__((ext_vector_type(8))) float v8f;
typedef int v4i_vs __attribute__((vector_size(16)));   // matches builtin param type

#define FDIM  128   // F_IN == H*C
#define HEADS 4
#define SLOPE 0.2f

#define GLOBAL_AS __attribute__((address_space(1)))
#define LOCAL_AS  __attribute__((address_space(3)))

#if defined(__has_builtin)
#  if __has_builtin(__builtin_amdgcn_global_load_async_to_lds_b128) && \
      __has_builtin(__builtin_amdgcn_s_wait_asynccnt)
#    define USE_ASYNC_LDS 1
#  endif
#endif
#ifndef USE_ASYNC_LDS
#  define USE_ASYNC_LDS 0
#endif

// ---- monotonic float<->uint encoding for atomicMax on floats ----
__device__ __forceinline__ unsigned enc_f32(float f) {
  unsigned u = __float_as_uint(f);
  return (u & 0x80000000u) ? ~u : (u | 0x80000000u);
}
__device__ __forceinline__ float dec_f32(unsigned e) {
  unsigned u = (e & 0x80000000u) ? (e & 0x7FFFFFFFu) : ~e;
  return __uint_as_float(u);
}

// ---------------------------------------------------------------------------
// 0) init: out = bias (broadcast), max = -inf (encoded), denom = 0
// ---------------------------------------------------------------------------
__global__ void gat_init(float* __restrict__ out, const float* __restrict__ bias,
                         unsigned* __restrict__ mmax, float* __restrict__ denom, int N) {
  long long idx = (long long)blockIdx.x * blockDim.x + threadIdx.x;
  long long total = (long long)N * FDIM;
  if (idx < total) out[idx] = bias[(int)(idx & (FDIM - 1))];
  if (idx < (long long)N * HEADS) {
    mmax[idx] = enc_f32(-__builtin_inff());
    denom[idx] = 0.0f;
  }
}

// ---------------------------------------------------------------------------
// 1) h = x @ W  via V_WMMA_F32_16X16X4_F32 (fp32-exact matrix core path)
//    block: 256 threads = 8 waves; 128 rows/block.
//    Dynamic LDS (128KB): sW = W with (k,k+1) pair interleave so B-frags are
//    single aligned ds_load_b64; sX = 128x128 x-tile staged via CDNA5
//    GLOBAL_LOAD_ASYNC_TO_LDS_B128 (ASYNCcnt) when available.
// ---------------------------------------------------------------------------
__global__ __launch_bounds__(256) void gat_gemm(const float* __restrict__ x,
                                                const float* __restrict__ W,
                                                float* __restrict__ h, int N) {
  extern __shared__ float smem[];
  float* sW = smem;                 // 16384 floats: sW[(k>>1)*256 + c*2 + (k&1)]
  float* sX = smem + FDIM * FDIM;   // 16384 floats: row-major 128x128 tile

  const int tid = threadIdx.x;
  const int blockRow = blockIdx.x * 128;

  // stage W with pair-interleaved layout (coalesced global reads)
  for (int i = tid; i < FDIM * FDIM; i += 256) {
    int k = i >> 7, c = i & 127;
    sW[((k >> 1) << 8) + (c << 1) + (k & 1)] = W[i];
  }

  // stage x tile (rows >= N clamped to N-1; writes guarded later)
#pragma unroll
  for (int jj = 0; jj < 16; ++jj) {
    int elt = (jj * 256 + tid) * 4;           // float index in tile (16B chunks)
    int r = elt >> 7, c = elt & 127;
    int gr = blockRow + r;
    if (gr >= N) gr = N - 1;
    const float* gsrc = x + (size_t)gr * FDIM + c;
    float* ldst = sX + elt;
#if USE_ASYNC_LDS
    __builtin_amdgcn_global_load_async_to_lds_b128(
        (GLOBAL_AS v4i_vs*)gsrc, (LOCAL_AS v4i_vs*)ldst, 0, 0);
#else
    *(float4*)ldst = *(const float4*)gsrc;
#endif
  }
#if USE_ASYNC_LDS
  __builtin_amdgcn_s_wait_asynccnt(0);
#endif
  __syncthreads();

  const int lane  = tid & 31;
  const int wave  = tid >> 5;
  const int laneM = lane & 15;          // M (A) / N (B) index within tile
  const int kHalf = (lane >> 4) << 1;   // lanes 0-15 -> K+0..1, lanes 16-31 -> K+2..3
  const float* sxrow = sX + (wave * 16 + laneM) * FDIM;

  v8f acc[8] = {};

  for (int k = 0; k < FDIM; k += 4) {
    const int kk = k + kHalf;                       // always even
    v2f a = *(const v2f*)(sxrow + kk);              // ds_load_b64
    const float* wpair = sW + ((kk >> 1) << 8);
#pragma unroll
    for (int t = 0; t < 8; ++t) {
      v2f b = *(const v2f*)(wpair + ((t * 16 + laneM) << 1));  // ds_load_b64
      acc[t] = __builtin_amdgcn_wmma_f32_16x16x4_f32(
          /*neg_a=*/false, a, /*neg_b=*/false, b,
          /*c_mod=*/(short)0, acc[t], /*reuse_a=*/false, /*reuse_b=*/false);
    }
  }

  // C/D layout: VGPR r holds M=r (lanes 0-15) and M=r+8 (lanes 16-31), N=lane&15
  const int rowBase = blockRow + wave * 16;
  const int mHi = (lane >> 4) << 3;
#pragma unroll
  for (int t = 0; t < 8; ++t) {
#pragma unroll
    for (int r = 0; r < 8; ++r) {
      int orow = rowBase + r + mHi;
      if (orow < N) h[(size_t)orow * FDIM + t * 16 + (lane & 15)] = acc[t][r];
    }
  }
}

// ---------------------------------------------------------------------------
// 2) a_src[n,h] = sum_c h[n,h,c]*att_src[h,c]; same for a_dst.
//    1 block per node; wave w == head w (C=32 == wave32); shuffle reduction.
// ---------------------------------------------------------------------------
__global__ __launch_bounds__(128) void gat_attcoef(const float* __restrict__ h,
                                                   const float* __restrict__ att_src,
                                                   const float* __restrict__ att_dst,
                                                   float* __restrict__ a_src,
                                                   float* __restrict__ a_dst, int N) {
  const int node = blockIdx.x;
  const int j = threadIdx.x;          // feature index 0..127
  float v = h[(size_t)node * FDIM + j];
  float s = v * att_src[j];
  float d = v * att_dst[j];
#pragma unroll
  for (int off = 16; off; off >>= 1) {
    s += __shfl_xor(s, off, 32);
    d += __shfl_xor(d, off, 32);
  }
  if ((j & 31) == 0) {
    const int head = j >> 5;
    a_src[node * HEADS + head] = s;
    a_dst[node * HEADS + head] = d;
  }
}

// ---------------------------------------------------------------------------
// 3) per-(edge,head) leaky-relu score, segment max over dst
// ---------------------------------------------------------------------------
__global__ void gat_edge_max(const long long* __restrict__ ei,
                             const float* __restrict__ a_src,
                             const float* __restrict__ a_dst,
                             unsigned* __restrict__ mmax, int E, int N) {
  long long idx = (long long)blockIdx.x * blockDim.x + threadIdx.x;
  long long total = (long long)(E + N) * HEADS;
  if (idx >= total) return;
  const int head = (int)(idx & 3);
  const long long edge = idx >> 2;
  int src, dst;
  if (edge < E) { src = (int)ei[edge]; dst = (int)ei[E + edge]; }
  else          { src = dst = (int)(edge - E); }          // self loop
  float e = a_src[src * HEADS + head] + a_dst[dst * HEADS + head];
  e = e > 0.0f ? e : SLOPE * e;
  atomicMax(&mmax[dst * HEADS + head], enc_f32(e));
}

// ---------------------------------------------------------------------------
// 4) denom[dst,h] += exp(e - max[dst,h])
// ---------------------------------------------------------------------------
__global__ void gat_edge_denom(const long long* __restrict__ ei,
                               const float* __restrict__ a_src,
                               const float* __restrict__ a_dst,
                               const unsigned* __restrict__ mmax,
                               float* __restrict__ denom, int E, int N) {
  long long idx = (long long)blockIdx.x * blockDim.x + threadIdx.x;
  long long total = (long long)(E + N) * HEADS;
  if (idx >= total) return;
  const int head = (int)(idx & 3);
  const long long edge = idx >> 2;
  int src, dst;
  if (edge < E) { src = (int)ei[edge]; dst = (int)ei[E + edge]; }
  else          { src = dst = (int)(edge - E); }
  float e = a_src[src * HEADS + head] + a_dst[dst * HEADS + head];
  e = e > 0.0f ? e : SLOPE * e;
  float p = __expf(e - dec_f32(mmax[dst * HEADS + head]));
  unsafeAtomicAdd(&denom[dst * HEADS + head], p);
}

// ---------------------------------------------------------------------------
// 5) out[dst] += alpha * h[src]  (128 threads per edge, one per feature;
//    alpha recomputed; h[src] row prefetched to overlap the alpha dep-chain)
// ---------------------------------------------------------------------------
__global__ __launch_bounds__(256) void gat_aggregate(const long long* __restrict__ ei,
                                                     const float* __restrict__ h,
                                                     const float* __restrict__ a_src,
                                                     const float* __restrict__ a_dst,
                                                     const unsigned* __restrict__ mmax,
                                                     const float* __restrict__ denom,
                                                     float* __restrict__ out, int E, int N) {
  long long t = (long long)blockIdx.x * blockDim.x + threadIdx.x;
  long long edge = t >> 7;
  if (edge >= (long long)(E + N)) return;
  const int j = (int)(t & 127);
  int src, dst;
  if (edge < E) { src = (int)ei[edge]; dst = (int)ei[E + edge]; }
  else          { src = dst = (int)(edge - E); }
  const float* hp = h + (size_t)src * FDIM + j;
  __builtin_prefetch(hp, 0, 1);                     // global_prefetch_b8
  const int head = j >> 5;
  float e = a_src[src * HEADS + head] + a_dst[dst * HEADS + head];
  e = e > 0.0f ? e : SLOPE * e;
  float alpha = __expf(e - dec_f32(mmax[dst * HEADS + head])) /
                (denom[dst * HEADS + head] + 1e-16f);
  unsafeAtomicAdd(&out[(size_t)dst * FDIM + j], alpha * *hp);
}

// ---------------------------------------------------------------------------
extern "C" void kernel_launch(void* const* d_in, const int* in_sizes, int n_in,
                              void* d_out, int out_size, void* d_ws, size_t ws_size,
                              hipStream_t stream) {
  const float*     x       = (const float*)d_in[0];
  const float*     W       = (const float*)d_in[1];
  const float*     att_src = (const float*)d_in[2];
  const float*     att_dst = (const float*)d_in[3];
  const float*     bias    = (const float*)d_in[4];
  const long long* ei      = (const long long*)d_in[5];  // int64 [2,E]

  const int N = in_sizes[0] / FDIM;
  const int E = in_sizes[5] / 2;
  float* out = (float*)d_out;

  // workspace layout: h[N*128] | a_src[N*4] | a_dst[N*4] | mmax[N*4] | denom[N*4]
  float*    h     = (float*)d_ws;
  float*    a_src = h + (size_t)N * FDIM;
  float*    a_dst = a_src + (size_t)N * HEADS;
  unsigned* mmax  = (unsigned*)(a_dst + (size_t)N * HEADS);
  float*    denom = (float*)(mmax + (size_t)N * HEADS);

  {
    long long tot = (long long)N * FDIM;
    gat_init<<<(int)((tot + 255) / 256), 256, 0, stream>>>(out, bias, mmax, denom, N);
  }
  gat_gemm<<<(N + 127) / 128, 256, 2 * FDIM * FDIM * sizeof(float), stream>>>(x, W, h, N);
  gat_attcoef<<<N, 128, 0, stream>>>(h, att_src, att_dst, a_src, a_dst, N);
  {
    long long tot = (long long)(E + N) * HEADS;
    int blocks = (int)((tot + 255) / 256);
    gat_edge_max<<<blocks, 256, 0, stream>>>(ei, a_src, a_dst, mmax, E, N);
    gat_edge_denom<<<blocks, 256, 0, stream>>>(ei, a_src, a_dst, mmax, denom, E, N);
  }
  {
    long long tot = (long long)(E + N) * FDIM;
    gat_aggregate<<<(int)((tot + 255) / 256), 256, 0, stream>>>(ei, h, a_src, a_dst,
                                                                mmax, denom, out, E, N);
  }
}